// SELayer_62569083568677
// MI455X (gfx1250) — compile-verified
//
#include <hip/hip_runtime.h>
#include <hip/hip_bf16.h>
#include <math.h>

// ---------------------------------------------------------------------------
// SE layer with channel pruning, MI455X (gfx1250, wave32).
//   x:[32,256,56,56] f32 -> pool -> MLP(256->16->256, relu/sigmoid)
//   -> top-k(128) per batch -> gather+scale -> out:[32,128,56,56]
//
// Roofline: ~205 MB total traffic => ~9 us floor @ 23.3 TB/s. MLP compute is
// negligible and done in full f32 precision on the WMMA pipe
// (V_WMMA_F32_16X16X4_F32). Top-k is parallelized one-workgroup-per-batch so
// its serial tail (2M pair-compares) stays well under the memory floor.
// ---------------------------------------------------------------------------

typedef __attribute__((ext_vector_type(2))) float v2f;
typedef __attribute__((ext_vector_type(8))) float v8f;

#define SE_B   32
#define SE_C   256
#define SE_CR  16      // C / R
#define SE_K   128     // channels kept
#define SE_HW  3136    // 56*56
#define SE_HW4 784     // float4 per channel

// ---------------------------------------------------------------------------
// Kernel 1: global average pool. One block per (b,c) channel. Streams 103 MB.
// ---------------------------------------------------------------------------
__global__ void __launch_bounds__(256) se_pool_kernel(const float* __restrict__ x,
                                                      float* __restrict__ s) {
  const int bc = blockIdx.x;  // b*256 + c, 0..8191
  const float4* __restrict__ src =
      reinterpret_cast<const float4*>(x + (size_t)bc * SE_HW);
  const int t = threadIdx.x;

  // 784 float4 = 3*256 + 16
  float4 v0 = src[t];
  float4 v1 = src[t + 256];
  float4 v2 = src[t + 512];
  float sum = (v0.x + v0.y + v0.z + v0.w) +
              (v1.x + v1.y + v1.z + v1.w) +
              (v2.x + v2.y + v2.z + v2.w);
  if (t < 16) {
    float4 v3 = src[t + 768];
    sum += v3.x + v3.y + v3.z + v3.w;
  }

  // wave32 reduction
  #pragma unroll
  for (int off = 16; off > 0; off >>= 1)
    sum += __shfl_down(sum, off, 32);

  __shared__ float red[8];
  if ((t & 31) == 0) red[t >> 5] = sum;
  __syncthreads();
  if (t == 0) {
    float tot = 0.f;
    #pragma unroll
    for (int i = 0; i < 8; ++i) tot += red[i];
    s[bc] = tot * (1.0f / (float)SE_HW);
  }
}

// ---------------------------------------------------------------------------
// Kernel 2: excitation MLP on the WMMA pipe (f32 16x16x4) + sigmoid.
// Single workgroup of 256 threads (8 waves). Writes G[32,256] to workspace.
//
// WMMA f32 16x16x4 layouts (ISA 7.12.2):
//   A (16x4, MxK):  lane l: m = l&15, half = l>>4; VGPR j holds K = 2*half+j
//   B (4x16,  KxN): lane l: n = l&15, half = l>>4; VGPR j holds K = 2*half+j
//   C/D (16x16):    lane l: n = l&15, half = l>>4; VGPR r holds M = r + 8*half
// ---------------------------------------------------------------------------
__global__ void __launch_bounds__(256) se_mlp_kernel(
    const float* __restrict__ s,    // [32,256]
    const float* __restrict__ w1,   // [16,256]  (out,in)
    const float* __restrict__ b1,   // [16]
    const float* __restrict__ w2,   // [256,16]  (out,in)
    const float* __restrict__ b2,   // [256]
    float* __restrict__ G) {        // [32,256] gates (workspace)

  __shared__ float lw1[SE_CR * SE_C];  // 16 KB
  __shared__ float lb1[SE_CR];
  __shared__ float lb2[SE_C];
  __shared__ float H[SE_B][SE_CR];     // 2 KB, post-ReLU hidden

  const int t = threadIdx.x;
  for (int i = t; i < SE_CR * SE_C; i += 256) lw1[i] = w1[i];
  if (t < SE_CR) lb1[t] = b1[t];
  lb2[t] = b2[t];
  __syncthreads();

  const int wave = t >> 5;
  const int lane = t & 31;
  const int half = lane >> 4;   // selects K sub-pair / M upper half
  const int ln   = lane & 15;   // M index for A, N index for B/C/D

  // ---- GEMM1: H = relu(S[32,256] @ W1^T[256,16] + b1). 2 M-tiles, K=256. ----
  if (wave < 2) {
    const int mt = wave;               // M-tile (batch rows 16*mt..16*mt+15)
    const int mA = ln + 16 * mt;       // A-matrix row for this lane
    v8f acc = {};
    #pragma unroll 4
    for (int k0 = 0; k0 < SE_C; k0 += 4) {
      const int kk = k0 + 2 * half;
      v2f a, b;
      a.x = s[mA * SE_C + kk];
      a.y = s[mA * SE_C + kk + 1];
      b.x = lw1[ln * SE_C + kk];       // W1^T[k][n] = w1[n*256 + k]
      b.y = lw1[ln * SE_C + kk + 1];
      acc = __builtin_amdgcn_wmma_f32_16x16x4_f32(
          /*neg_a=*/false, a, /*neg_b=*/false, b,
          /*c_mod=*/(short)0, acc, /*reuse_a=*/false, /*reuse_b=*/false);
    }
    #pragma unroll
    for (int r = 0; r < 8; ++r) {
      const int m = r + 8 * half + 16 * mt;
      const float h = acc[r] + lb1[ln];
      H[m][ln] = h > 0.f ? h : 0.f;
    }
  }
  __syncthreads();

  // ---- GEMM2: G = sigmoid(H[32,16] @ W2^T[16,256] + b2). 2x16 tiles. ----
  for (int tt = wave * 4; tt < wave * 4 + 4; ++tt) {  // 4 tiles per wave
    const int mt = tt >> 4;
    const int nt = tt & 15;
    const int n  = nt * 16 + ln;       // global output channel
    const int mA = ln + 16 * mt;
    v8f acc = {};
    #pragma unroll
    for (int k0 = 0; k0 < SE_CR; k0 += 4) {
      const int kk = k0 + 2 * half;
      v2f a, b;
      a.x = H[mA][kk];
      a.y = H[mA][kk + 1];
      b.x = w2[n * SE_CR + kk];        // W2^T[k][n] = w2[n*16 + k]
      b.y = w2[n * SE_CR + kk + 1];
      acc = __builtin_amdgcn_wmma_f32_16x16x4_f32(
          false, a, false, b, (short)0, acc, false, false);
    }
    #pragma unroll
    for (int r = 0; r < 8; ++r) {
      const int m = r + 8 * half + 16 * mt;
      const float gv = acc[r] + lb2[n];
      G[m * SE_C + n] = 1.0f / (1.0f + __expf(-gv));
    }
  }
}

// ---------------------------------------------------------------------------
// Kernel 3: top-k(128) per batch via stable rank counting. One block per
// batch (32 blocks); matches lax.top_k order exactly (descending values,
// ties broken by ascending index). G row lives in LDS; the inner loop is a
// conflict-free LDS broadcast read.
// ---------------------------------------------------------------------------
__global__ void __launch_bounds__(256) se_topk_kernel(
    const float* __restrict__ G,     // [32,256]
    float* __restrict__ sel_gate,    // [32,128]
    int*   __restrict__ sel_idx) {   // [32,128]
  __shared__ float row[SE_C];
  const int b = blockIdx.x;
  const int t = threadIdx.x;         // channel
  row[t] = G[b * SE_C + t];
  __syncthreads();

  const float g = row[t];
  int rank = 0;
  #pragma unroll 16
  for (int c2 = 0; c2 < SE_C; ++c2) {
    const float g2 = row[c2];
    rank += (g2 > g) || (g2 == g && c2 < t);
  }
  if (rank < SE_K) {
    sel_gate[b * SE_K + rank] = g;
    sel_idx[b * SE_K + rank]  = t;
  }
}

// ---------------------------------------------------------------------------
// Kernel 4: gather selected channels and scale. One block per (b,k).
// Reads hit L2 (x fits in the 192 MB global L2 after the pool pass).
// ---------------------------------------------------------------------------
__global__ void __launch_bounds__(256) se_gather_kernel(
    const float* __restrict__ x,
    const float* __restrict__ sel_gate,
    const int*   __restrict__ sel_idx,
    float* __restrict__ out) {
  const int bk = blockIdx.x;          // b*128 + k
  const int b  = bk >> 7;
  const float gate = sel_gate[bk];
  const int   c    = sel_idx[bk];

  const float4* __restrict__ src =
      reinterpret_cast<const float4*>(x + ((size_t)b * SE_C + c) * SE_HW);
  float4* __restrict__ dst =
      reinterpret_cast<float4*>(out + (size_t)bk * SE_HW);

  for (int i = threadIdx.x; i < SE_HW4; i += 256) {
    if (i + 256 < SE_HW4) __builtin_prefetch(&src[i + 256], 0, 1);
    float4 v = src[i];
    float4 o;
    o.x = v.x * gate; o.y = v.y * gate; o.z = v.z * gate; o.w = v.w * gate;
    dst[i] = o;
  }
}

// ---------------------------------------------------------------------------
// Launch
// ---------------------------------------------------------------------------
extern "C" void kernel_launch(void* const* d_in, const int* in_sizes, int n_in,
                              void* d_out, int out_size, void* d_ws, size_t ws_size,
                              hipStream_t stream) {
  const float* x  = (const float*)d_in[0];   // [32,256,56,56]
  const float* w1 = (const float*)d_in[1];   // [16,256]
  const float* b1 = (const float*)d_in[2];   // [16]
  const float* w2 = (const float*)d_in[3];   // [256,16]
  const float* b2 = (const float*)d_in[4];   // [256]
  float* out = (float*)d_out;                // [32,128,56,56]

  // workspace: s[8192] f32 | G[8192] f32 | sel_gate[4096] f32 | sel_idx[4096] i32
  float* s        = (float*)d_ws;
  float* G        = s + SE_B * SE_C;
  float* sel_gate = G + SE_B * SE_C;
  int*   sel_idx  = (int*)(sel_gate + SE_B * SE_K);

  se_pool_kernel<<<SE_B * SE_C, 256, 0, stream>>>(x, s);
  se_mlp_kernel<<<1, 256, 0, stream>>>(s, w1, b1, w2, b2, G);
  se_topk_kernel<<<SE_B, 256, 0, stream>>>(G, sel_gate, sel_idx);
  se_gather_kernel<<<SE_B * SE_K, 256, 0, stream>>>(x, sel_gate, sel_idx, out);
}